// OptimizationModel_52407190946155
// MI455X (gfx1250) — compile-verified
//
#include <hip/hip_runtime.h>
#include <math.h>

#define N_OBJ 8
#define N_PTS 2048
#define K_NN  10
#define TOL_CONTACT 0.01f
#define FLT_BIG 3.4e38f

typedef float v2f __attribute__((ext_vector_type(2)));
typedef float v8f __attribute__((ext_vector_type(8)));

// ---------------------------------------------------------------------------
// 4x4 inverse (adjugate method). Works on flat arrays independent of major
// order since inv(M^T) = inv(M)^T.
// ---------------------------------------------------------------------------
__device__ __forceinline__ void invert4x4(const float* m, float* invOut) {
  float inv[16];
  inv[0]  =  m[5]*m[10]*m[15] - m[5]*m[11]*m[14] - m[9]*m[6]*m[15] + m[9]*m[7]*m[14] + m[13]*m[6]*m[11] - m[13]*m[7]*m[10];
  inv[4]  = -m[4]*m[10]*m[15] + m[4]*m[11]*m[14] + m[8]*m[6]*m[15] - m[8]*m[7]*m[14] - m[12]*m[6]*m[11] + m[12]*m[7]*m[10];
  inv[8]  =  m[4]*m[9]*m[15]  - m[4]*m[11]*m[13] - m[8]*m[5]*m[15] + m[8]*m[7]*m[13] + m[12]*m[5]*m[11] - m[12]*m[7]*m[9];
  inv[12] = -m[4]*m[9]*m[14]  + m[4]*m[10]*m[13] + m[8]*m[5]*m[14] - m[8]*m[6]*m[13] - m[12]*m[5]*m[10] + m[12]*m[6]*m[9];
  inv[1]  = -m[1]*m[10]*m[15] + m[1]*m[11]*m[14] + m[9]*m[2]*m[15] - m[9]*m[3]*m[14] - m[13]*m[2]*m[11] + m[13]*m[3]*m[10];
  inv[5]  =  m[0]*m[10]*m[15] - m[0]*m[11]*m[14] - m[8]*m[2]*m[15] + m[8]*m[3]*m[14] + m[12]*m[2]*m[11] - m[12]*m[3]*m[10];
  inv[9]  = -m[0]*m[9]*m[15]  + m[0]*m[11]*m[13] + m[8]*m[1]*m[15] - m[8]*m[3]*m[13] - m[12]*m[1]*m[11] + m[12]*m[3]*m[9];
  inv[13] =  m[0]*m[9]*m[14]  - m[0]*m[10]*m[13] - m[8]*m[1]*m[14] + m[8]*m[2]*m[13] + m[12]*m[1]*m[10] - m[12]*m[2]*m[9];
  inv[2]  =  m[1]*m[6]*m[15]  - m[1]*m[7]*m[14]  - m[5]*m[2]*m[15] + m[5]*m[3]*m[14] + m[13]*m[2]*m[7]  - m[13]*m[3]*m[6];
  inv[6]  = -m[0]*m[6]*m[15]  + m[0]*m[7]*m[14]  + m[4]*m[2]*m[15] - m[4]*m[3]*m[14] - m[12]*m[2]*m[7]  + m[12]*m[3]*m[6];
  inv[10] =  m[0]*m[5]*m[15]  - m[0]*m[7]*m[13]  - m[4]*m[1]*m[15] + m[4]*m[3]*m[13] + m[12]*m[1]*m[7]  - m[12]*m[3]*m[5];
  inv[14] = -m[0]*m[5]*m[14]  + m[0]*m[6]*m[13]  + m[4]*m[1]*m[14] - m[4]*m[2]*m[13] - m[12]*m[1]*m[6]  + m[12]*m[2]*m[5];
  inv[3]  = -m[1]*m[6]*m[11]  + m[1]*m[7]*m[10]  + m[5]*m[2]*m[11] - m[5]*m[3]*m[10] - m[9]*m[2]*m[7]   + m[9]*m[3]*m[6];
  inv[7]  =  m[0]*m[6]*m[11]  - m[0]*m[7]*m[10]  - m[4]*m[2]*m[11] + m[4]*m[3]*m[10] + m[8]*m[2]*m[7]   - m[8]*m[3]*m[6];
  inv[11] = -m[0]*m[5]*m[11]  + m[0]*m[7]*m[9]   + m[4]*m[1]*m[11] - m[4]*m[3]*m[9]  - m[8]*m[1]*m[7]   + m[8]*m[3]*m[5];
  inv[15] =  m[0]*m[5]*m[10]  - m[0]*m[6]*m[9]   - m[4]*m[1]*m[10] + m[4]*m[2]*m[9]  + m[8]*m[1]*m[6]   - m[8]*m[2]*m[5];
  float det = m[0]*inv[0] + m[1]*inv[4] + m[2]*inv[8] + m[3]*inv[12];
  det = 1.0f / det;
  #pragma unroll
  for (int i = 0; i < 16; i++) invOut[i] = inv[i] * det;
}

// ---------------------------------------------------------------------------
// Kernel 1: per-object transform. X[n][p] = M[n]*xyz (+t), padded to float4
// with w = 0 so a K=4 WMMA dot equals the 3-D dot. SQ[n][p] = |x|^2.
// ---------------------------------------------------------------------------
__global__ __launch_bounds__(256) void prep_kernel(const float* __restrict__ pts,
                                                   const float* __restrict__ T_est,
                                                   const float* __restrict__ T_plane,
                                                   float4* __restrict__ X,
                                                   float*  __restrict__ SQ) {
  const int n = blockIdx.x;
  __shared__ float M[16];
  if (threadIdx.x == 0) {
    float invP[16];
    invert4x4(T_plane, invP);
    #pragma unroll
    for (int r = 0; r < 4; ++r)
      #pragma unroll
      for (int c = 0; c < 4; ++c) {
        float acc = 0.0f;
        #pragma unroll
        for (int k = 0; k < 4; ++k) acc += invP[r*4 + k] * T_est[n*16 + k*4 + c];
        M[r*4 + c] = acc;
      }
  }
  __syncthreads();
  for (int p = threadIdx.x; p < N_PTS; p += 256) {
    const float* pt = pts + ((size_t)n * N_PTS + p) * 6;
    float px = pt[0], py = pt[1], pz = pt[2];
    float x = M[0]*px + M[1]*py + M[2]*pz  + M[3];
    float y = M[4]*px + M[5]*py + M[6]*pz  + M[7];
    float z = M[8]*px + M[9]*py + M[10]*pz + M[11];
    X[(size_t)n * N_PTS + p] = make_float4(x, y, z, 0.0f);
    SQ[(size_t)n * N_PTS + p] = x*x + y*y + z*z;
  }
}

// ---------------------------------------------------------------------------
// Kernel 2: one block per (p-tile, b, o). Stages object o into LDS, then each
// wave computes 32x16 dot tiles with v_wmma_f32_16x16x4_f32, bounces the C
// tiles through a PER-WAVE LDS patch so each lane owns one p row, and keeps a
// register-resident sorted top-10 in the shifted domain c = |xq|^2 - 2*dot
// (the per-lane constant |xp|^2 cannot change the ordering), so a rejected
// candidate costs just one v_fmac + one v_cmpx. Same-wave LDS ops are
// in-order on CDNA5, so only a compiler code-motion fence (wave_barrier) is
// needed around the tile exchange -- no s_barrier traffic in the hot loop.
// ---------------------------------------------------------------------------
__global__ __launch_bounds__(256) void pair_kernel(const float*  __restrict__ pts,
                                                   const float4* __restrict__ X,
                                                   const float*  __restrict__ SQ,
                                                   float* __restrict__ D0) {
  const int ptile = blockIdx.x;   // 0..7  (256 p's each)
  const int b     = blockIdx.y;   // 0..7
  const int o     = blockIdx.z;   // 0..7
  if (b == o) return;             // diagonal is masked to +inf downstream

  __shared__ float4 sX[N_PTS];          // 32 KB: object o coords (w = 0)
  __shared__ float  sSQ[N_PTS];         // 8 KB : object o |x|^2
  __shared__ float  sTile[8][32 * 16];  // 16 KB: per-wave WMMA dot tiles

  const int tid = threadIdx.x;
  for (int q = tid; q < N_PTS; q += 256) {
    sX[q]  = X[(size_t)o * N_PTS + q];
    sSQ[q] = SQ[(size_t)o * N_PTS + q];
  }
  __syncthreads();

  const int wave    = tid >> 5;
  const int lane    = tid & 31;
  const int halfsel = lane >> 4;        // 0: carries K0,K1   1: carries K2,K3
  const int sub     = lane & 15;
  const int pbase   = ptile * 256 + wave * 32;
  const int p       = pbase + lane;     // this lane's point in object b

  // A fragments (ISA 7.12.2, 32-bit A 16x4): lanes 0-15 hold (K0,K1) of row M,
  // lanes 16-31 hold (K2,K3) of row M.  Two fragments: rows p0..15 / p16..31.
  float4 xa0 = X[(size_t)b * N_PTS + pbase + sub];
  float4 xa1 = X[(size_t)b * N_PTS + pbase + 16 + sub];
  v2f a_lo, a_hi;
  a_lo[0] = halfsel ? xa0.z : xa0.x;  a_lo[1] = halfsel ? xa0.w : xa0.y;
  a_hi[0] = halfsel ? xa1.z : xa1.x;  a_hi[1] = halfsel ? xa1.w : xa1.y;

  const float4 xown  = X[(size_t)b * N_PTS + p];
  const float  sqown = SQ[(size_t)b * N_PTS + p];

  // Top-10 in shifted domain: bd[] holds c = |xq|^2 - 2*dot (d2 - sqown).
  float bd[K_NN];
  int   bi[K_NN];
  #pragma unroll
  for (int t = 0; t < K_NN; ++t) { bd[t] = FLT_BIG; bi[t] = 0; }

  float* myTile = &sTile[wave][0];
  const float4* myRow = (const float4*)&sTile[wave][lane * 16];  // 64B aligned
  const int col    = sub;
  const int rowoff = halfsel * 8;

  for (int q0 = 0; q0 < N_PTS; q0 += 16) {
    // B fragment: 4x16 (KxN), column N = q0+sub, mirrored K split across halves
    float4 xb = sX[q0 + sub];
    v2f bf;
    bf[0] = halfsel ? xb.z : xb.x;
    bf[1] = halfsel ? xb.w : xb.y;

    v8f c0 = {};
    v8f c1 = {};
    c0 = __builtin_amdgcn_wmma_f32_16x16x4_f32(false, a_lo, false, bf, (short)0, c0, false, false);
    c1 = __builtin_amdgcn_wmma_f32_16x16x4_f32(false, a_hi, false, bf, (short)0, c1, false, false);

    // Scatter C tiles (VGPR v, lane l -> row v + 8*(l>>4), col l&15) to LDS.
    #pragma unroll
    for (int v = 0; v < 8; ++v) {
      myTile[(rowoff + v) * 16 + col]      = c0[v];
      myTile[(16 + rowoff + v) * 16 + col] = c1[v];
    }
    // Per-wave exchange: HW keeps same-wave LDS ops in order; fence only the
    // compiler so the row reads are not hoisted above the scatter stores.
    __builtin_amdgcn_wave_barrier();

    // Each lane consumes its own 64-byte p-row as 4 x b128 loads; the 16
    // |xq|^2 values are uniform-address b128 broadcasts.
    float4 r0 = myRow[0];
    float4 r1 = myRow[1];
    float4 r2 = myRow[2];
    float4 r3 = myRow[3];
    const float4* sq4 = (const float4*)(sSQ + q0);   // 64B aligned, uniform
    float4 s0 = sq4[0];
    float4 s1 = sq4[1];
    float4 s2 = sq4[2];
    float4 s3 = sq4[3];
    float dots[16] = { r0.x, r0.y, r0.z, r0.w,  r1.x, r1.y, r1.z, r1.w,
                       r2.x, r2.y, r2.z, r2.w,  r3.x, r3.y, r3.z, r3.w };
    float sqs[16]  = { s0.x, s0.y, s0.z, s0.w,  s1.x, s1.y, s1.z, s1.w,
                       s2.x, s2.y, s2.z, s2.w,  s3.x, s3.y, s3.z, s3.w };
    #pragma unroll
    for (int j = 0; j < 16; ++j) {
      float cv = fmaf(-2.0f, dots[j], sqs[j]);   // shifted d2 (constant sqown dropped)
      if (cv < bd[K_NN - 1]) {                   // cheap reject for ~99% of candidates
        int qidx = q0 + j;
        #pragma unroll
        for (int t = K_NN - 1; t >= 1; --t) {
          bool up   = cv < bd[t - 1];
          bool here = (!up) && (cv < bd[t]);
          bd[t] = up ? bd[t - 1] : (here ? cv   : bd[t]);
          bi[t] = up ? bi[t - 1] : (here ? qidx : bi[t]);
        }
        if (cv < bd[0]) { bd[0] = cv; bi[0] = qidx; }
      }
    }
    // Keep next iteration's scatter stores after this iteration's reads.
    __builtin_amdgcn_wave_barrier();
  }

  // Normal vote: sign(dot(n_q, grad)) is invariant under grad normalization.
  int cnt = 0;
  #pragma unroll
  for (int t = 0; t < K_NN; ++t) {
    int q = bi[t];
    float4 nq = sX[q];
    float gx = nq.x - xown.x, gy = nq.y - xown.y, gz = nq.z - xown.z;
    const float* np = pts + ((size_t)o * N_PTS + q) * 6 + 3;
    float dn = np[0]*gx + np[1]*gy + np[2]*gz;
    cnt += (dn > 0.0f) ? 1 : 0;
  }
  bool inside = cnt > 8;                         // sum > k*0.8 = 8
  float d0 = sqrtf(fmaxf(sqown + bd[0], 0.0f));  // un-shift nearest distance
  D0[((size_t)b * N_OBJ + o) * N_PTS + p] = inside ? -d0 : d0;
}

// ---------------------------------------------------------------------------
// Kernel 3: sd_other = min over o != b; signed = min(z, sd_other); outputs.
// ---------------------------------------------------------------------------
__global__ __launch_bounds__(256) void finalize_kernel(const float4* __restrict__ X,
                                                       const float*  __restrict__ D0,
                                                       float* __restrict__ out) {
  int i = blockIdx.x * blockDim.x + threadIdx.x;
  if (i >= N_OBJ * N_PTS) return;
  int b = i / N_PTS;
  int p = i - b * N_PTS;
  float z = X[i].z;
  float sd = FLT_BIG;
  #pragma unroll
  for (int o = 0; o < N_OBJ; ++o) {
    if (o == b) continue;
    sd = fminf(sd, D0[((size_t)b * N_OBJ + o) * N_PTS + p]);
  }
  float s = fminf(z, sd);
  out[i] = s;
  out[N_OBJ * N_PTS + i] = (s < -TOL_CONTACT) ? 1.0f : 0.0f;
}

// ---------------------------------------------------------------------------
extern "C" void kernel_launch(void* const* d_in, const int* in_sizes, int n_in,
                              void* d_out, int out_size, void* d_ws, size_t ws_size,
                              hipStream_t stream) {
  const float* pts     = (const float*)d_in[0];  // (8,2048,6)
  const float* T_est   = (const float*)d_in[1];  // (8,4,4)
  const float* T_plane = (const float*)d_in[2];  // (4,4)
  // d_in[3] = k (always 10; hardcoded)

  float* ws = (float*)d_ws;
  float4* X  = (float4*)ws;                          // 8*2048 float4 = 256 KB
  float*  SQ = ws + (size_t)N_OBJ * N_PTS * 4;       // 8*2048 floats =  64 KB
  float*  D0 = SQ + (size_t)N_OBJ * N_PTS;           // 8*8*2048      = 512 KB

  prep_kernel<<<N_OBJ, 256, 0, stream>>>(pts, T_est, T_plane, X, SQ);

  dim3 grid(N_PTS / 256, N_OBJ, N_OBJ);
  pair_kernel<<<grid, 256, 0, stream>>>(pts, X, SQ, D0);

  finalize_kernel<<<(N_OBJ * N_PTS + 255) / 256, 256, 0, stream>>>(X, D0, (float*)d_out);
}